// Combiner_52158082842993
// MI455X (gfx1250) — compile-verified
//
#include <hip/hip_runtime.h>
#include <hip/hip_bf16.h>

// ---------------- problem constants (from reference) ----------------
constexpr int BS  = 16;
constexpr int SEQ = 4096;
constexpr int DIM = 512;          // floats per row (2 KB)
constexpr int META_THREADS = 1024;
constexpr int META_CHUNKS  = (BS * SEQ) / META_THREADS;   // 64

typedef float v4f __attribute__((ext_vector_type(4)));
typedef int   gv4i __attribute__((vector_size(16)));      // matches builtin param

#define AS1 __attribute__((address_space(1)))
#define AS3 __attribute__((address_space(3)))

// CDNA5 async global<->LDS data mover path (gfx1250 only), guarded so the
// host pass and any toolchain lacking the builtins still compile.
#if defined(__gfx1250__) && defined(__has_builtin)
#if __has_builtin(__builtin_amdgcn_global_load_async_to_lds_b128) && \
    __has_builtin(__builtin_amdgcn_global_store_async_from_lds_b128)
#define USE_ASYNC_LDS 1
#endif
#endif
#ifndef USE_ASYNC_LDS
#define USE_ASYNC_LDS 0
#endif

__device__ __forceinline__ void wait_asynccnt0() {
#if defined(__gfx1250__)
#if defined(__has_builtin) && __has_builtin(__builtin_amdgcn_s_wait_asynccnt)
  __builtin_amdgcn_s_wait_asynccnt(0);
#else
  asm volatile("s_wait_asynccnt 0" ::: "memory");
#endif
#endif
}

// ---------------------------------------------------------------------
// Phase 1: one workgroup scans all BS*SEQ labels in row-major order.
// Produces, per output slot (b,p):
//   packed = l(12b) | is_mask<<12 | is_end<<13 | rep_row<<14   (rep_row: 16b)
// plus new_len[b] (workspace) and the float lens output.
// ---------------------------------------------------------------------
__global__ void combiner_meta_kernel(const int* __restrict__ labels,
                                     int* __restrict__ ws_len,
                                     int* __restrict__ ws_slot,
                                     float* __restrict__ out_lens) {
  __shared__ int s[META_THREADS];
  const int tid = threadIdx.x;
  int runKeep = 0;   // per-batch running kept count (uniform across threads)
  int runEnd  = 0;   // global running END count      (uniform across threads)

  for (int c = 0; c < META_CHUNKS; ++c) {
    const int gi  = c * META_THREADS + tid;      // global row-major index
    const int b   = gi >> 12;                    // / SEQ
    const int lab = labels[gi];
    const int keep = (lab != 1) ? 1 : 0;         // SKIPPED|END|MASKED survive
    const int e    = (lab == 2) ? 1 : 0;         // SUBWORD_END
    const int m    = (lab == 3) ? 1 : 0;         // MASKED_TOKEN

    s[tid] = keep | (e << 16);                   // packed dual scan
    __syncthreads();
    for (int off = 1; off < META_THREADS; off <<= 1) {
      int v = (tid >= off) ? s[tid - off] : 0;
      __syncthreads();
      s[tid] += v;
      __syncthreads();
    }
    const int incl = s[tid];
    const int tot  = s[META_THREADS - 1];

    if (keep) {
      const int kpos = runKeep + (incl & 0xffff) - 1;       // output position
      const int ridx = runEnd + (incl >> 16) - 1;           // global END idx
      const int l    = gi & (SEQ - 1);
      const int sv   = l | (m << 12) | (e << 13) | (e ? (ridx << 14) : 0);
      ws_slot[b * SEQ + kpos] = sv;
    }

    runKeep += tot & 0xffff;
    runEnd  += tot >> 16;
    if ((c & 3) == 3) {                          // end of one batch row
      if (tid == 0) {
        ws_len[b]   = runKeep;
        out_lens[b] = (float)runKeep;
      }
      runKeep = 0;
    }
    __syncthreads();                             // protect s[] for next chunk
  }
}

// ---------------------------------------------------------------------
// Phase 2: gather. One block per output row (b,p); 128 threads, 16B/thread.
// Kept rows are staged through LDS via the async data mover (ASYNCcnt).
// ---------------------------------------------------------------------
__global__ void combiner_gather_kernel(const float* __restrict__ encoded,
                                       const float* __restrict__ representation,
                                       const int* __restrict__ ws_len,
                                       const int* __restrict__ ws_slot,
                                       float* __restrict__ out) {
  __shared__ __align__(16) float buf[DIM];       // 2 KB staging tile

  const int tid = threadIdx.x;                   // 0..127
  const int bid = blockIdx.x;                    // 0..BS*SEQ-1
  const int b   = bid >> 12;
  const int p   = bid & (SEQ - 1);

  float* const out_enc  = out;
  float* const out_mask = out + (size_t)BS * SEQ * DIM + BS;

  const size_t row = (size_t)b * SEQ + p;
  float* const dst = out_enc + row * DIM;

  const int nlen = ws_len[b];
  if (p >= nlen) {                               // uniform per block
    v4f z = {0.f, 0.f, 0.f, 0.f};
    ((v4f*)dst)[tid] = z;
    if (tid == 0) out_mask[row] = 0.0f;
    return;
  }

  const int sv   = ws_slot[row];
  const int l    = sv & 0xfff;
  const int m    = (sv >> 12) & 1;
  const int e    = (sv >> 13) & 1;
  const int ridx = (sv >> 14) & 0xffff;

  const float* src = e ? (representation + (size_t)ridx * DIM)
                       : (encoded + ((size_t)b * SEQ + l) * DIM);

#if USE_ASYNC_LDS
  // Generic-AS typed pointers first, then a pure address-space cast.
  gv4i* sgen = (gv4i*)(src + tid * 4);           // drops const, retypes pointee
  gv4i* lgen = (gv4i*)(buf + tid * 4);
  gv4i* dgen = (gv4i*)(dst + tid * 4);

  // global -> LDS (async, ASYNCcnt), per-lane 16B chunk
  __builtin_amdgcn_global_load_async_to_lds_b128(
      (AS1 gv4i*)sgen, (AS3 gv4i*)lgen, 0, 0);
  wait_asynccnt0();                              // our lane's LDS data landed
  // LDS -> global (async, ASYNCcnt)
  __builtin_amdgcn_global_store_async_from_lds_b128(
      (AS1 gv4i*)dgen, (AS3 gv4i*)lgen, 0, 0);
  wait_asynccnt0();
#else
  ((v4f*)dst)[tid] = ((const v4f*)src)[tid];
  (void)buf;
#endif

  if (tid == 0) out_mask[row] = m ? 1.0f : 0.0f;
}

// ---------------------------------------------------------------------
extern "C" void kernel_launch(void* const* d_in, const int* in_sizes, int n_in,
                              void* d_out, int out_size, void* d_ws, size_t ws_size,
                              hipStream_t stream) {
  const float* encoded        = (const float*)d_in[0];   // [16,4096,512] f32
  const float* representation = (const float*)d_in[1];   // [65536,512]   f32
  const int*   labels         = (const int*)d_in[2];     // [16,4096]
  // d_in[3] = lengths: unused by the reference outputs

  float* out = (float*)d_out;
  float* out_lens = out + (size_t)BS * SEQ * DIM;        // 16 floats

  int* ws_len  = (int*)d_ws;                             // 16 ints
  int* ws_slot = ws_len + 16;                            // 65536 ints

  combiner_meta_kernel<<<1, META_THREADS, 0, stream>>>(labels, ws_len, ws_slot,
                                                       out_lens);
  combiner_gather_kernel<<<BS * SEQ, DIM / 4, 0, stream>>>(
      encoded, representation, ws_len, ws_slot, out);
}